// TopographicalCorticalCell_8194797601536
// MI455X (gfx1250) — compile-verified
//
#include <hip/hip_runtime.h>

#ifndef __has_builtin
#define __has_builtin(x) 0
#endif

#if __has_builtin(__builtin_amdgcn_global_load_async_to_lds_b128)
#define USE_ASYNC_B128 1
#else
#define USE_ASYNC_B128 0
#endif
#if __has_builtin(__builtin_amdgcn_global_load_async_to_lds_b32)
#define USE_ASYNC_B32 1
#else
#define USE_ASYNC_B32 0
#endif

// Geometry (matches reference constants)
#define R_DIM 256
#define C_DIM 256
#define S_SYN 32               // random synapses per neuron
#define EPN   (S_SYN + 1)      // +1 self synapse = 33 edges per source neuron
#define B_SZ  64               // batch
#define N_SZ  (R_DIM * C_DIM)  // 65536 neurons

typedef int v4i __attribute__((ext_vector_type(4)));
typedef __attribute__((address_space(1))) int* gptr_i32;
typedef __attribute__((address_space(3))) int* lptr_i32;
typedef __attribute__((address_space(1))) v4i* gptr_v4i;
typedef __attribute__((address_space(3))) v4i* lptr_v4i;

__device__ __forceinline__ void wait_async0() {
#if __has_builtin(__builtin_amdgcn_s_wait_asynccnt)
  __builtin_amdgcn_s_wait_asynccnt(0);
#else
  asm volatile("s_wait_asynccnt 0" ::: "memory");
#endif
}

// ---------------------------------------------------------------------------
// Relaxed agent-scope f32 atomic add -> global_atomic_add_f32 (no return)
// ---------------------------------------------------------------------------
__device__ __forceinline__ void atom_add_f32(float* p, float v) {
  (void)__hip_atomic_fetch_add(p, v, __ATOMIC_RELAXED, __HIP_MEMORY_SCOPE_AGENT);
}

// ---------------------------------------------------------------------------
// 64x64-tiled transpose with compile-time dims: out[c, r] = in[r, c].
// in  : [IN_ROWS, IN_COLS] row-major ; out : [IN_COLS, IN_ROWS] row-major
// grid = (IN_COLS/64, IN_ROWS/64), block = 256 (8 wave32)
// Load phase: CDNA5 async data mover, b128 (16 B/lane), ASYNCcnt-tracked.
// Tile pitch 68 floats = 272 B keeps every row 16 B aligned for b128.
// Store phase: float4 stores, fully coalesced (256 B per 16 lanes).
// ---------------------------------------------------------------------------
template <int IN_ROWS, int IN_COLS>
__global__ __launch_bounds__(256)
void transpose64(const float* __restrict__ in, float* __restrict__ out)
{
  __shared__ float tile[64][68];
  const int c0 = blockIdx.x * 64;
  const int r0 = blockIdx.y * 64;
  const int t  = threadIdx.x;

  // ---- load phase: rows of `in` -> LDS tile -------------------------------
  const int c4 = (t & 15) * 4;   // 4 consecutive cols per lane
  const int rA = t >> 4;         // 0..15, step 16 x4
#if USE_ASYNC_B128
  #pragma unroll
  for (int i = 0; i < 4; ++i) {
    const int r = rA + i * 16;
    const float* src = in + (size_t)(r0 + r) * IN_COLS + (c0 + c4);
    __builtin_amdgcn_global_load_async_to_lds_b128(
        (gptr_v4i)src, (lptr_v4i)&tile[r][c4], /*offset=*/0, /*cpol=*/0);
  }
  wait_async0();
#elif USE_ASYNC_B32
  #pragma unroll
  for (int i = 0; i < 4; ++i) {
    const int r = rA + i * 16;
    const float* src = in + (size_t)(r0 + r) * IN_COLS + (c0 + c4);
    #pragma unroll
    for (int j = 0; j < 4; ++j)
      __builtin_amdgcn_global_load_async_to_lds_b32(
          (gptr_i32)(src + j), (lptr_i32)&tile[r][c4 + j], 0, 0);
  }
  wait_async0();
#else
  #pragma unroll
  for (int i = 0; i < 4; ++i) {
    const int r = rA + i * 16;
    const float4 v = *(const float4*)(in + (size_t)(r0 + r) * IN_COLS + (c0 + c4));
    *(float4*)&tile[r][c4] = v;
  }
#endif
  __syncthreads();

  // ---- store phase: LDS tile columns -> rows of `out` ---------------------
  const int or4 = (t & 15) * 4;  // 4 consecutive out-fast-dim elems per lane
  const int ocA = t >> 4;        // 0..15, step 16 x4
  #pragma unroll
  for (int i = 0; i < 4; ++i) {
    const int c = ocA + i * 16;
    float4 v;
    v.x = tile[or4 + 0][c];
    v.y = tile[or4 + 1][c];
    v.z = tile[or4 + 2][c];
    v.w = tile[or4 + 3][c];
    *(float4*)(out + (size_t)(c0 + c) * IN_ROWS + (r0 + or4)) = v;
  }
}

// ---------------------------------------------------------------------------
// Scatter SpMM: one wave32 per source neuron n (edges grouped 33-per-column).
// Lane l owns batch elements 2l, 2l+1 (float2 of xt[n]).
// The 32 random edges are preloaded one-per-lane (coalesced) and broadcast
// with v_readlane (constant lane after unroll) -> row/value land in SGPRs,
// so the 64 contiguous global_atomic_add_f32 per edge use SADDR addressing.
// xt, out_t are [N, 64] row-major.  block = 256 threads = 8 neurons.
// ---------------------------------------------------------------------------
__global__ __launch_bounds__(256)
void spmm_scatter(const float* __restrict__ xt,
                  const float* __restrict__ values,
                  const int*   __restrict__ rows,
                  const int*   __restrict__ cols,
                  float*       __restrict__ out_t)
{
  const int lane = threadIdx.x & 31;
  const int wave = threadIdx.x >> 5;
  const int n    = blockIdx.x * 8 + wave;
  const size_t base = (size_t)n * EPN;

  const int c = cols[base];  // wave-uniform: all 33 edges share the source col
  const float2 xv = *(const float2*)(xt + (size_t)c * B_SZ + lane * 2);

  const int      r_l = rows[base + lane];
  const unsigned v_l = __float_as_uint(values[base + lane]);

  #pragma unroll
  for (int k = 0; k < 32; ++k) {
    const int   rr = __builtin_amdgcn_readlane(r_l, k);
    const float vv = __uint_as_float(__builtin_amdgcn_readlane(v_l, k));
    float* dst = out_t + (size_t)rr * B_SZ + lane * 2;
    atom_add_f32(dst,     vv * xv.x);
    atom_add_f32(dst + 1, vv * xv.y);
  }

  // 33rd edge: self synapse (uniform loads)
  const int   rs = rows[base + S_SYN];
  const float vs = values[base + S_SYN];
  float* dst = out_t + (size_t)rs * B_SZ + lane * 2;
  atom_add_f32(dst,     vs * xv.x);
  atom_add_f32(dst + 1, vs * xv.y);
}

// ---------------------------------------------------------------------------
// Fallback (only if workspace is too small for the two [N,64] buffers):
// same scatter but strided directly on x [B,N] / out [B,N].
// ---------------------------------------------------------------------------
__global__ __launch_bounds__(256)
void spmm_scatter_strided(const float* __restrict__ x,
                          const float* __restrict__ values,
                          const int*   __restrict__ rows,
                          const int*   __restrict__ cols,
                          float*       __restrict__ out)
{
  const int lane = threadIdx.x & 31;
  const int wave = threadIdx.x >> 5;
  const int n    = blockIdx.x * 8 + wave;
  const size_t base = (size_t)n * EPN;

  const int c = cols[base];
  const float x0 = x[(size_t)lane        * N_SZ + c];
  const float x1 = x[(size_t)(lane + 32) * N_SZ + c];

  const int      r_l = rows[base + lane];
  const unsigned v_l = __float_as_uint(values[base + lane]);

  #pragma unroll
  for (int k = 0; k < 32; ++k) {
    const int   rr = __builtin_amdgcn_readlane(r_l, k);
    const float vv = __uint_as_float(__builtin_amdgcn_readlane(v_l, k));
    atom_add_f32(out + (size_t)lane        * N_SZ + rr, vv * x0);
    atom_add_f32(out + (size_t)(lane + 32) * N_SZ + rr, vv * x1);
  }
  const int   rs = rows[base + S_SYN];
  const float vs = values[base + S_SYN];
  atom_add_f32(out + (size_t)lane        * N_SZ + rs, vs * x0);
  atom_add_f32(out + (size_t)(lane + 32) * N_SZ + rs, vs * x1);
}

// ---------------------------------------------------------------------------
extern "C" void kernel_launch(void* const* d_in, const int* in_sizes, int n_in,
                              void* d_out, int out_size, void* d_ws, size_t ws_size,
                              hipStream_t stream)
{
  const float* x      = (const float*)d_in[0];
  const float* values = (const float*)d_in[1];
  const int*   rows   = (const int*)d_in[2];
  const int*   cols   = (const int*)d_in[3];
  float* out = (float*)d_out;

  const size_t buf_bytes = (size_t)N_SZ * B_SZ * sizeof(float);  // 16 MB each

  if (ws_size >= 2 * buf_bytes) {
    float* xt    = (float*)d_ws;                          // [N, 64]
    float* out_t = (float*)((char*)d_ws + buf_bytes);     // [N, 64]

    (void)hipMemsetAsync(out_t, 0, buf_bytes, stream);
    // x [64, N] -> xt [N, 64]
    transpose64<B_SZ, N_SZ><<<dim3(N_SZ / 64, B_SZ / 64), 256, 0, stream>>>(x, xt);
    // scatter: out_t[r,:] += v * xt[c,:]
    spmm_scatter<<<N_SZ / 8, 256, 0, stream>>>(xt, values, rows, cols, out_t);
    // out_t [N, 64] -> out [64, N]
    transpose64<N_SZ, B_SZ><<<dim3(B_SZ / 64, N_SZ / 64), 256, 0, stream>>>(out_t, out);
  } else {
    // Workspace-free fallback: strided accesses, still correct.
    (void)hipMemsetAsync(out, 0, (size_t)out_size * sizeof(float), stream);
    spmm_scatter_strided<<<N_SZ / 8, 256, 0, stream>>>(x, values, rows, cols, out);
  }
}